// DownSampler_2465311228438
// MI455X (gfx1250) — compile-verified
//
#include <hip/hip_runtime.h>

// DownSampler: y = decimate4( conv1d_same(x, W) ), W[o,i,k] = b[k] for all o,i.
// => out[n,o,u] = sum_j b[j] * S[n, 4u + j - 40], S[n,t] = sum_c x[n,c,t],
//    identical across all 64 output channels. Memory-bound (~82 MB traffic,
//    ~3.5 us at 23.3 TB/s).
//
// Pass 1: channel sum (reads 64 MB, writes 1 MB scratch).
// Pass 2: decimating FIR as fp32 WMMA GEMM (v_wmma_f32_16x16x4_f32).
//   Y[m,n] = sum_k A[m,k]*B[k,n], A[m,k] = S[t0+64m+k], B[k,n] = b[k-4n].
//   A window and pre-swizzled B fragments are staged in LDS so the 36-chunk
//   K-loop is branch-free: 2x ds_load_b64 + 1x wmma per chunk, fully unrolled.

typedef __attribute__((ext_vector_type(2))) float v2f;
typedef __attribute__((ext_vector_type(8))) float v8f;

#define B_SZ    8
#define C_SZ    64
#define T_IN    32768
#define T_OUT   8192
#define NTAPS   81
#define PADW    40
#define TILE    256      // decimated outputs per WMMA tile (16x16)
#define KCHUNKS 36       // K = 144 = 36 chunks of 4 (need k up to 80 + 4*15 = 140)
#define SWIN    1152     // staged S window: 64*15 + 144 = 1104, rounded to 4.5*256
#define SWIN4   (SWIN / 4)

__global__ void __launch_bounds__(256) ds_chansum(const float* __restrict__ x,
                                                  float* __restrict__ S) {
  int tid = blockIdx.x * 256 + threadIdx.x;     // 65536 threads total
  int n   = tid >> 13;                          // batch
  int t4  = tid & 8191;                         // float4 index along time
  const float4* xp = (const float4*)x + (size_t)n * (C_SZ * (T_IN / 4)) + t4;
  float4 acc = make_float4(0.f, 0.f, 0.f, 0.f);
#pragma unroll 8
  for (int c = 0; c < C_SZ; ++c) {
    float4 v = xp[(size_t)c * (T_IN / 4)];
    acc.x += v.x; acc.y += v.y; acc.z += v.z; acc.w += v.w;
  }
  ((float4*)S)[(size_t)n * (T_IN / 4) + t4] = acc;
}

__global__ void __launch_bounds__(256) ds_fir_wmma(const float* __restrict__ S,
                                                   const float* __restrict__ weight,
                                                   float* __restrict__ out) {
  __shared__ __align__(16) float sS[SWIN];            // zero-padded S window
  __shared__ __align__(16) float sB[KCHUNKS * 64];    // B frags: [kc][lane][v]
  __shared__ __align__(16) float sD[TILE];            // GEMM result tile

  int tid   = threadIdx.x;
  int tile  = blockIdx.x;           // 256 tiles = 8 batches * 32 tiles
  int n     = tile >> 5;
  int obase = (tile & 31) * TILE;   // first decimated output of this tile
  int t0    = obase * 4 - PADW;     // S-domain window base (multiple of 4, may be <0)
  const float* Sn = S + (size_t)n * T_IN;

  // --- Stage pre-swizzled B fragments: sB[kc*64 + lane*2 + v] = b[k - 4*(lane&15)],
  //     k = 4*kc + 2*(lane>>4) + v.  2304 entries / 256 threads = 9 each.
  for (int idx = tid; idx < KCHUNKS * 64; idx += 256) {
    int kc   = idx >> 6;
    int lane = (idx >> 1) & 31;
    int v    = idx & 1;
    int k    = 4 * kc + 2 * (lane >> 4) + v;
    int j    = k - 4 * (lane & 15);
    sB[idx]  = ((unsigned)j < (unsigned)NTAPS) ? weight[j] : 0.0f;
  }

  // --- Stage S window with zero-padded halo (vector path for interior).
  for (int q = tid; q < SWIN4; q += 256) {
    int t = t0 + 4 * q;
    float4 v;
    if (t >= 0 && t + 3 < T_IN) {
      v = *(const float4*)(Sn + t);
    } else {
      v.x = ((unsigned)(t + 0) < (unsigned)T_IN) ? Sn[t + 0] : 0.0f;
      v.y = ((unsigned)(t + 1) < (unsigned)T_IN) ? Sn[t + 1] : 0.0f;
      v.z = ((unsigned)(t + 2) < (unsigned)T_IN) ? Sn[t + 2] : 0.0f;
      v.w = ((unsigned)(t + 3) < (unsigned)T_IN) ? Sn[t + 3] : 0.0f;
    }
    ((float4*)sS)[q] = v;
  }
  __syncthreads();

  // --- Wave 0: 16x16x144 fp32 GEMM on the matrix pipe (EXEC all-ones).
  if (tid < 32) {
    int h = tid >> 4;               // lane half (0/1) -> K parity pair
    int l = tid & 15;               // M (for A) / N (for B) index
    const float* ap = sS + 64 * l + 2 * h;   // A[m=l, k=4kc+2h+v]: consecutive pair
    const float* bp = sB + tid * 2;          // pre-swizzled pair for this lane

    v8f acc = {0.f, 0.f, 0.f, 0.f, 0.f, 0.f, 0.f, 0.f};
#pragma unroll
    for (int kc = 0; kc < KCHUNKS; ++kc) {
      v2f a = *(const v2f*)(ap + 4 * kc);    // ds_load_b64, 8B aligned
      v2f b = *(const v2f*)(bp + 64 * kc);   // ds_load_b64, 8B aligned
      acc = __builtin_amdgcn_wmma_f32_16x16x4_f32(
          /*neg_a=*/false, a, /*neg_b=*/false, b,
          /*c_mod=*/(short)0, acc, /*reuse_a=*/false, /*reuse_b=*/false);
    }
    // D layout: VGPR r, lane -> M = r + 8*h, N = l ; Y[m,n] = out[obase+16m+n]
#pragma unroll
    for (int r = 0; r < 8; ++r)
      sD[16 * (r + 8 * h) + l] = acc[r];
  }
  __syncthreads();

  // --- Broadcast the 256-sample tile to all 64 identical output channels.
  int q = tid & 63;                             // float4 index within tile
  int cbase = tid >> 6;                         // 0..3
  float4 val = ((const float4*)sD)[q];
  float4* op = (float4*)out + (size_t)n * C_SZ * (T_OUT / 4) + (obase >> 2) + q;
#pragma unroll
  for (int i = 0; i < 16; ++i) {
    int c = cbase * 16 + i;
    op[(size_t)c * (T_OUT / 4)] = val;          // coalesced 1KB bursts
  }
}

extern "C" void kernel_launch(void* const* d_in, const int* in_sizes, int n_in,
                              void* d_out, int out_size, void* d_ws, size_t ws_size,
                              hipStream_t stream) {
  const float* x = (const float*)d_in[0];       // [8, 64, 32768] fp32
  const float* w = (const float*)d_in[1];       // [64, 64, 81]  fp32 (b = w[0:81])
  float* out = (float*)d_out;                   // [8, 64, 8192] fp32
  float* S   = (float*)d_ws;                    // scratch: 8*32768 fp32 = 1 MB

  ds_chansum<<<256, 256, 0, stream>>>(x, S);
  ds_fir_wmma<<<256, 256, 0, stream>>>(S, w, out);
}